// GraphAttentionDense_11433202942856
// MI455X (gfx1250) — compile-verified
//
#include <hip/hip_runtime.h>
#include <hip/hip_bf16.h>
#include <float.h>

typedef float v2f __attribute__((ext_vector_type(2)));
typedef float v8f __attribute__((ext_vector_type(8)));

#define WMMA_F32(a, b, c) \
    __builtin_amdgcn_wmma_f32_16x16x4_f32(false, (a), false, (b), (short)0, (c), false, false)

static constexpr int B_    = 2;
static constexpr int N_    = 2048;
static constexpr int C_    = 256;
static constexpr int H_    = 8;
static constexpr int D_    = 64;
static constexpr int INNER = 512;   // H_*D_
static constexpr int QKVW  = 1536;  // 3*INNER
static constexpr int KNN_  = 32;
static constexpr float SCALE_ = 0.125f; // 64^-0.5

// ---------------------------------------------------------------------------
// K0: x2[row] = sum_c x[row][c]^2   (one wave per row, 8 waves/block)
// ---------------------------------------------------------------------------
__global__ void k_rowsumsq(const float* __restrict__ x, float* __restrict__ x2) {
    int wave = threadIdx.x >> 5;
    int lane = threadIdx.x & 31;
    int row  = blockIdx.x * 8 + wave;           // [0, B_*N_)
    const float* xr = x + (size_t)row * C_;
    float s = 0.f;
    #pragma unroll
    for (int i = lane; i < C_; i += 32) { float v = xr[i]; s += v * v; }
    #pragma unroll
    for (int off = 16; off; off >>= 1) s += __shfl_xor(s, off, 32);
    if (lane == 0) x2[row] = s;
}

// ---------------------------------------------------------------------------
// K1: d2[b][i][j] = x2[i]+x2[j]-2*dot(x_i,x_j), diag=FLT_MAX.
// WMMA f32 16x16x4; one wave per 16x16 output tile, 8 waves/block.
// grid = (N_/16/8, N_/16, B_)
// ---------------------------------------------------------------------------
__global__ void k_gram_d2(const float* __restrict__ x, const float* __restrict__ x2,
                          float* __restrict__ d2) {
    int wave = threadIdx.x >> 5;
    int lane = threadIdx.x & 31;
    int half = lane >> 4;
    int l    = lane & 15;
    int i0   = (blockIdx.x * 8 + wave) * 16;
    int j0   = blockIdx.y * 16;
    int b    = blockIdx.z;
    const float* xb   = x + (size_t)b * N_ * C_;
    const float* arow = xb + (size_t)(i0 + l) * C_ + 2 * half;  // A: M=l (per half), K pair
    const float* brow = xb + (size_t)(j0 + l) * C_ + 2 * half;  // B = X^T: N=l, same striping
    v8f c = {};
    #pragma unroll 4
    for (int k = 0; k < C_; k += 4) {
        v2f a, bb;
        a.x  = arow[k];  a.y  = arow[k + 1];
        bb.x = brow[k];  bb.y = brow[k + 1];
        c = WMMA_F32(a, bb, c);
    }
    const float* x2b = x2 + (size_t)b * N_;
    #pragma unroll
    for (int r = 0; r < 8; ++r) {
        int i = i0 + r + 8 * half;   // C/D layout: VGPR r -> M = r (+8 for upper half)
        int j = j0 + l;              // N = lane & 15
        float v = x2b[i] + x2b[j] - 2.0f * c[r];
        if (i == j) v = FLT_MAX;     // exclude self-loop
        d2[((size_t)b * N_ + i) * N_ + j] = v;
    }
}

// ---------------------------------------------------------------------------
// K2: per-row top-32 smallest (lowest-index tie-break, matching top_k).
// One 128-thread block per row; row cached in LDS; 32x argmin extraction.
// ---------------------------------------------------------------------------
__global__ void k_top32(const float* __restrict__ d2, int* __restrict__ knn) {
    __shared__ float sv[N_];
    __shared__ float rv[128];
    __shared__ int   ri[128];
    int row = blockIdx.x;                        // b*N_ + n
    int t   = threadIdx.x;
    const float* dr = d2 + (size_t)row * N_;
    for (int i = t; i < N_; i += 128) sv[i] = dr[i];
    __syncthreads();
    int* out = knn + row * KNN_;
    for (int s = 0; s < KNN_; ++s) {
        float best = INFINITY;
        int   bi   = N_;
        for (int i = t; i < N_; i += 128) {      // ascending scan -> first min = lowest idx
            float v = sv[i];
            if (v < best) { best = v; bi = i; }
        }
        rv[t] = best; ri[t] = bi;
        __syncthreads();
        for (int off = 64; off; off >>= 1) {
            if (t < off) {
                if (rv[t + off] < rv[t] ||
                    (rv[t + off] == rv[t] && ri[t + off] < ri[t])) {
                    rv[t] = rv[t + off]; ri[t] = ri[t + off];
                }
            }
            __syncthreads();
        }
        int m = ri[0];
        if (t == 0) { out[s] = m; sv[m] = INFINITY; }
        __syncthreads();
    }
}

// ---------------------------------------------------------------------------
// K3: qkv = x @ Wqkv  ([B_*N_,256] @ [256,1536]).  WMMA f32 16x16x4.
// grid = (M/16/8, 1536/16)
// ---------------------------------------------------------------------------
__global__ void k_gemm_qkv(const float* __restrict__ x, const float* __restrict__ W,
                           float* __restrict__ qkv) {
    int wave = threadIdx.x >> 5;
    int lane = threadIdx.x & 31;
    int half = lane >> 4;
    int l    = lane & 15;
    int i0   = (blockIdx.x * 8 + wave) * 16;
    int j0   = blockIdx.y * 16;
    const float* arow = x + (size_t)(i0 + l) * C_ + 2 * half;
    const float* bcol = W + (size_t)(2 * half) * QKVW + (j0 + l);
    v8f c = {};
    #pragma unroll 4
    for (int k = 0; k < C_; k += 4) {
        v2f a, bb;
        a.x  = arow[k];                    a.y  = arow[k + 1];
        bb.x = bcol[(size_t)k * QKVW];     bb.y = bcol[(size_t)(k + 1) * QKVW];
        c = WMMA_F32(a, bb, c);
    }
    #pragma unroll
    for (int r = 0; r < 8; ++r)
        qkv[(size_t)(i0 + r + 8 * half) * QKVW + j0 + l] = c[r];
}

// ---------------------------------------------------------------------------
// K4: sparse 32-neighbor attention. One wave32 per (b,n,h): lane = neighbor
// slot for QK^T + softmax (shfl reductions), then lane = 2 output dims for PV.
// Block 256 = 8 waves = all 8 heads of one (b,n); grid = B_*N_.
// ---------------------------------------------------------------------------
__global__ void k_attn(const float* __restrict__ qkv, const int* __restrict__ knn,
                       float* __restrict__ attn) {
    int h    = threadIdx.x >> 5;
    int lane = threadIdx.x & 31;
    int row  = blockIdx.x;               // b*N_ + n
    int b    = row >> 11;                // /N_
    int m    = knn[row * KNN_ + lane];   // this lane's neighbor index
    const float* q  = qkv + (size_t)row * QKVW + h * D_;
    const float* kk = qkv + (size_t)(b * N_ + m) * QKVW + INNER + h * D_;
    float dot = 0.f;
    #pragma unroll
    for (int d = 0; d < D_; ++d) dot += q[d] * kk[d];
    dot *= SCALE_;
    // softmax over the 32 lanes
    float mx = dot;
    #pragma unroll
    for (int off = 16; off; off >>= 1) mx = fmaxf(mx, __shfl_xor(mx, off, 32));
    float e = expf(dot - mx);
    float se = e;
    #pragma unroll
    for (int off = 16; off; off >>= 1) se += __shfl_xor(se, off, 32);
    float p = e / se;
    // out[d] = sum_s p_s * v[m_s][d] ; each lane owns 2 dims
    float a0 = 0.f, a1 = 0.f;
    #pragma unroll 4
    for (int s = 0; s < KNN_; ++s) {
        float ps = __shfl(p, s, 32);
        int   ms = __shfl(m, s, 32);
        const float* v = qkv + (size_t)(b * N_ + ms) * QKVW + 2 * INNER + h * D_;
        a0 += ps * v[lane * 2];
        a1 += ps * v[lane * 2 + 1];
    }
    float* o = attn + (size_t)row * INNER + h * D_;
    o[lane * 2]     = a0;
    o[lane * 2 + 1] = a1;
}

// ---------------------------------------------------------------------------
// K5: out = attn @ Wout + bout ([4096,512] @ [512,256]). Bias folded into C.
// grid = (M/16/8, 256/16)
// ---------------------------------------------------------------------------
__global__ void k_gemm_out(const float* __restrict__ attn, const float* __restrict__ W,
                           const float* __restrict__ bias, float* __restrict__ out) {
    int wave = threadIdx.x >> 5;
    int lane = threadIdx.x & 31;
    int half = lane >> 4;
    int l    = lane & 15;
    int i0   = (blockIdx.x * 8 + wave) * 16;
    int j0   = blockIdx.y * 16;
    const float* arow = attn + (size_t)(i0 + l) * INNER + 2 * half;
    const float* bcol = W + (size_t)(2 * half) * C_ + (j0 + l);
    float bj = bias[j0 + l];
    v8f c = {bj, bj, bj, bj, bj, bj, bj, bj};
    #pragma unroll 4
    for (int k = 0; k < INNER; k += 4) {
        v2f a, bb;
        a.x  = arow[k];                 a.y  = arow[k + 1];
        bb.x = bcol[(size_t)k * C_];    bb.y = bcol[(size_t)(k + 1) * C_];
        c = WMMA_F32(a, bb, c);
    }
    #pragma unroll
    for (int r = 0; r < 8; ++r)
        out[(size_t)(i0 + r + 8 * half) * C_ + j0 + l] = c[r];
}

// ---------------------------------------------------------------------------
extern "C" void kernel_launch(void* const* d_in, const int* in_sizes, int n_in,
                              void* d_out, int out_size, void* d_ws, size_t ws_size,
                              hipStream_t stream) {
    const float* x    = (const float*)d_in[0];   // [B,N,C]
    const float* Wqkv = (const float*)d_in[1];   // [C, 3*inner]
    const float* Wout = (const float*)d_in[2];   // [inner, C]
    const float* bout = (const float*)d_in[3];   // [C]
    float* out = (float*)d_out;                  // [B,N,C]

    // workspace layout (floats)
    float* ws   = (float*)d_ws;
    float* x2   = ws;                                      // B*N            = 4096
    float* d2   = x2 + (size_t)B_ * N_;                    // B*N*N          = 8388608
    float* qkv  = d2 + (size_t)B_ * N_ * N_;               // B*N*1536       = 6291456
    float* attn = qkv + (size_t)B_ * N_ * QKVW;            // B*N*512        = 2097152
    int*   knn  = (int*)(attn + (size_t)B_ * N_ * INNER);  // B*N*32         = 131072

    const int MROWS = B_ * N_;  // 4096

    k_rowsumsq<<<MROWS / 8, 256, 0, stream>>>(x, x2);
    k_gram_d2<<<dim3(N_ / 16 / 8, N_ / 16, B_), 256, 0, stream>>>(x, x2, d2);
    k_top32<<<MROWS, 128, 0, stream>>>(d2, knn);
    k_gemm_qkv<<<dim3(MROWS / 16 / 8, QKVW / 16), 256, 0, stream>>>(x, Wqkv, qkv);
    k_attn<<<MROWS, 256, 0, stream>>>(qkv, knn, attn);
    k_gemm_out<<<dim3(MROWS / 16 / 8, C_ / 16), 256, 0, stream>>>(attn, Wout, bout, out);
}